// TemporalRetentionWalker_45071386804738
// MI455X (gfx1250) — compile-verified
//
#include <hip/hip_runtime.h>
#include <hip/hip_bf16.h>
#include <math.h>

// Problem constants from the reference: B=4, S=128, T=2048, D=2048
constexpr int Bc = 4;
constexpr int Sc = 128;
constexpr int Tc = 2048;
constexpr int Dc = 2048;
constexpr int NCHUNK = 64;          // temporal chunks for the kv reduction
constexpr int CH = Tc / NCHUNK;     // 32 timesteps per chunk
constexpr float LN_EPS = 1e-5f;

typedef __attribute__((ext_vector_type(2))) float v2f;
typedef __attribute__((ext_vector_type(8))) float v8f;

// ---------------------------------------------------------------------------
// Kernel 1: weighted temporal partial sums of kv.
//   partial[b, c, d] = sum_{t in chunk c} w_t * kv[b, t, d]
//   w_t = (1 - K^{T-t}) / (1 - K),  K = sigmoid(decay)
// grid: (D/1024, NCHUNK, B), block 256; each thread handles 4 consecutive d
// (float4 coalesced loads -> full-bandwidth streaming of the 64 MB kv tensor).
// ---------------------------------------------------------------------------
__global__ __launch_bounds__(256) void k_weighted_partial(
    const float* __restrict__ kv, const float* __restrict__ decay,
    float* __restrict__ partial) {
  const int tid = threadIdx.x;
  const int d4 = (blockIdx.x * 256 + tid) * 4;
  const int c = blockIdx.y;
  const int b = blockIdx.z;

  const float K = 1.0f / (1.0f + expf(-decay[0]));
  const float invOm = 1.0f / (1.0f - K);

  const int t0 = c * CH;
  const int t1 = t0 + CH;
  // iterate t descending so the running power only multiplies by K (stable)
  float q = powf(K, (float)(Tc - (t1 - 1)));  // K^{T-t} at t = t1-1 (underflow->0 ok)

  float4 acc = make_float4(0.f, 0.f, 0.f, 0.f);
  const float* base = kv + (size_t)b * Tc * Dc + d4;
  for (int t = t1 - 1; t >= t0; --t) {
    const float w = (1.0f - q) * invOm;
    const float4 v = *reinterpret_cast<const float4*>(base + (size_t)t * Dc);
    acc.x += w * v.x;
    acc.y += w * v.y;
    acc.z += w * v.z;
    acc.w += w * v.w;
    q *= K;
  }
  *reinterpret_cast<float4*>(partial + (size_t)(b * NCHUNK + c) * Dc + d4) = acc;
}

// ---------------------------------------------------------------------------
// Kernel 2: fold chunk partials -> kvw16 [16, D] (rows 4..15 zeroed so the
// WMMA GEMM can run with a fully dense, divergence-free 16-row A matrix).
// grid: (D/1024, 16), block 256.
// ---------------------------------------------------------------------------
__global__ __launch_bounds__(256) void k_reduce(
    const float* __restrict__ partial, float* __restrict__ kvw16) {
  const int tid = threadIdx.x;
  const int d4 = (blockIdx.x * 256 + tid) * 4;
  const int b = blockIdx.y;  // 0..15
  float4 acc = make_float4(0.f, 0.f, 0.f, 0.f);
  if (b < Bc) {
    for (int c = 0; c < NCHUNK; ++c) {
      const float4 v = *reinterpret_cast<const float4*>(
          partial + (size_t)(b * NCHUNK + c) * Dc + d4);
      acc.x += v.x;
      acc.y += v.y;
      acc.z += v.z;
      acc.w += v.w;
    }
  }
  *reinterpret_cast<float4*>(kvw16 + (size_t)b * Dc + d4) = acc;
}

// ---------------------------------------------------------------------------
// Kernel 3/4: C[m, e] = (sum_d A[m, d] * W[e, d] + biasScale * bias[e]) * outScale
// A is [16, D] (rows >= 4 are zero / don't-care), W is row-major [e][d].
// One block (8 waves) per 16-wide e-tile; each wave owns a K-slice of 256 and
// issues V_WMMA_F32_16X16X4_F32 (fp32 precision); partial C tiles reduced in LDS.
//
// WMMA f32 16x16x4 lane layouts (ISA 7.12.2):
//   A: lane -> M = lane&15, VGPR{0,1} = K = (lane>>4)*2 + {0,1}
//   B: lane -> N = lane&15, VGPR{0,1} = K = (lane>>4)*2 + {0,1}
//   C: VGPR j -> M = j + (lane>=16 ? 8 : 0), N = lane&15
// mode==1: biasScale = sum_t w_t, outScale = 1/T (the proj_in path)
// mode==0: biasScale = 1, outScale = 1       (the proj_out path)
// ---------------------------------------------------------------------------
__global__ __launch_bounds__(256) void k_gemm16(
    const float* __restrict__ A, const float* __restrict__ W,
    const float* __restrict__ bias, const float* __restrict__ decay, int mode,
    float* __restrict__ out) {
  __shared__ float lds[8][32][8];

  const int tid = threadIdx.x;
  const int lane = tid & 31;
  const int wv = tid >> 5;
  const int n = lane & 15;
  const int koff = (lane >> 4) * 2;
  const int ebase = blockIdx.x * 16;

  const int KSPAN = Dc / 8;  // 256 d-values per wave
  const int dstart = wv * KSPAN;

  const float* arow = A + (size_t)n * Dc;            // A row (M = n for this lane)
  const float* wrow = W + (size_t)(ebase + n) * Dc;  // W row (N = n for this lane)

  v8f c0 = {0.f, 0.f, 0.f, 0.f, 0.f, 0.f, 0.f, 0.f};
  v8f c1 = {0.f, 0.f, 0.f, 0.f, 0.f, 0.f, 0.f, 0.f};

#pragma unroll 4
  for (int d = dstart; d < dstart + KSPAN; d += 8) {
    const int k0 = d + koff;
    const v2f a0 = *reinterpret_cast<const v2f*>(arow + k0);
    const v2f b0 = *reinterpret_cast<const v2f*>(wrow + k0);
    const v2f a1 = *reinterpret_cast<const v2f*>(arow + k0 + 4);
    const v2f b1 = *reinterpret_cast<const v2f*>(wrow + k0 + 4);
    c0 = __builtin_amdgcn_wmma_f32_16x16x4_f32(false, a0, false, b0, (short)0,
                                               c0, false, false);
    c1 = __builtin_amdgcn_wmma_f32_16x16x4_f32(false, a1, false, b1, (short)0,
                                               c1, false, false);
  }

  const v8f cs = c0 + c1;
#pragma unroll
  for (int j = 0; j < 8; ++j) lds[wv][lane][j] = cs[j];
  __syncthreads();

  // cross-wave K reduction: thread (lf, j) owns C element (M, N)
  const int lf = tid & 31;
  const int j = tid >> 5;
  float s = 0.f;
#pragma unroll
  for (int w = 0; w < 8; ++w) s += lds[w][lf][j];

  const int M = j + ((lf >= 16) ? 8 : 0);
  const int N = lf & 15;

  float biasScale, outScale;
  if (mode == 1) {
    const float K = 1.0f / (1.0f + expf(-decay[0]));
    const float om = 1.0f - K;
    const float kT = powf(K, (float)Tc);  // ~0 for K=sigmoid(0.9)
    const float sumw = ((float)Tc - K * (1.0f - kT) / om) / om;
    biasScale = sumw;
    outScale = 1.0f / (float)Tc;
  } else {
    biasScale = 1.0f;
    outScale = 1.0f;
  }
  // rows M >= 4 hold don't-care data; they are never consumed downstream
  out[(size_t)M * Dc + ebase + N] = (s + biasScale * bias[ebase + N]) * outScale;
}

// ---------------------------------------------------------------------------
// Kernel 5: y = Z[b,s,:] + summary[b,:]; LayerNorm over D (exact two-pass
// mean / mean((y-mu)^2), matching the reference numerics).
// grid: B*S blocks, 256 threads; 8 elements per thread held in registers.
// ---------------------------------------------------------------------------
__device__ __forceinline__ float block_reduce_sum(float v, float* red) {
#pragma unroll
  for (int o = 16; o > 0; o >>= 1) v += __shfl_xor(v, o, 32);
  const int lane = threadIdx.x & 31;
  const int wv = threadIdx.x >> 5;
  if (lane == 0) red[wv] = v;
  __syncthreads();
  float t = 0.f;
#pragma unroll
  for (int w = 0; w < 8; ++w) t += red[w];
  __syncthreads();
  return t;
}

__global__ __launch_bounds__(256) void k_ln(
    const float* __restrict__ Z, const float* __restrict__ summary,
    const float* __restrict__ gamma, const float* __restrict__ beta,
    float* __restrict__ out) {
  __shared__ float red[8];
  const int row = blockIdx.x;       // 0 .. B*S-1
  const int b = row >> 7;           // S = 128
  const int tid = threadIdx.x;

  const float* zrow = Z + (size_t)row * Dc;
  const float* srow = summary + (size_t)b * Dc;

  float y[8];
  float lsum = 0.f;
#pragma unroll
  for (int i = 0; i < 8; ++i) {
    const int idx = tid + i * 256;
    y[i] = zrow[idx] + srow[idx];
    lsum += y[i];
  }
  const float mu = block_reduce_sum(lsum, red) * (1.0f / (float)Dc);

  float lvar = 0.f;
#pragma unroll
  for (int i = 0; i < 8; ++i) {
    const float d = y[i] - mu;
    lvar += d * d;
  }
  const float var = block_reduce_sum(lvar, red) * (1.0f / (float)Dc);
  const float rstd = rsqrtf(var + LN_EPS);

  float* orow = out + (size_t)row * Dc;
#pragma unroll
  for (int i = 0; i < 8; ++i) {
    const int idx = tid + i * 256;
    orow[idx] = (y[i] - mu) * rstd * gamma[idx] + beta[idx];
  }
}

// ---------------------------------------------------------------------------
extern "C" void kernel_launch(void* const* d_in, const int* in_sizes, int n_in,
                              void* d_out, int out_size, void* d_ws,
                              size_t ws_size, hipStream_t stream) {
  const float* Z = (const float*)d_in[0];      // [B,S,D]
  const float* kv = (const float*)d_in[1];     // [B,T,D]
  const float* decay = (const float*)d_in[2];  // scalar
  const float* W_in = (const float*)d_in[3];   // [D,D]
  const float* b_in = (const float*)d_in[4];   // [D]
  const float* W_out = (const float*)d_in[5];  // [D,D]
  const float* b_out = (const float*)d_in[6];  // [D]
  const float* gamma = (const float*)d_in[7];  // [D]
  const float* beta = (const float*)d_in[8];   // [D]
  float* out = (float*)d_out;

  float* ws = (float*)d_ws;
  float* partial = ws;                              // NCHUNK*B*D = 524288 floats
  float* kvw16 = partial + (size_t)NCHUNK * Bc * Dc; // 16*D
  float* Ms16 = kvw16 + 16 * Dc;                     // 16*D
  float* sum16 = Ms16 + 16 * Dc;                     // 16*D
  (void)in_sizes; (void)n_in; (void)out_size; (void)ws_size;

  // 1) weighted temporal reduction of kv (bandwidth-dominant: 64 MB)
  k_weighted_partial<<<dim3(Dc / 1024, NCHUNK, Bc), 256, 0, stream>>>(
      kv, decay, partial);
  // 2) fold partials into padded 16-row A matrix
  k_reduce<<<dim3(Dc / 1024, 16), 256, 0, stream>>>(partial, kvw16);
  // 3) mean_states = (kvw @ W_in^T + sumw*b_in) / T   (fp32 WMMA)
  k_gemm16<<<Dc / 16, 256, 0, stream>>>(kvw16, W_in, b_in, decay, 1, Ms16);
  // 4) summary = mean_states @ W_out^T + b_out        (fp32 WMMA)
  k_gemm16<<<Dc / 16, 256, 0, stream>>>(Ms16, W_out, b_out, decay, 0, sum16);
  // 5) broadcast add + LayerNorm
  k_ln<<<Bc * Sc, 256, 0, stream>>>(Z, sum16, gamma, beta, out);
}